// PlasticCortex_48704929137327
// MI455X (gfx1250) — compile-verified
//
#include <hip/hip_runtime.h>
#include <math.h>

#define B_ 128
#define S_ 8192
#define E_ 32
#define H_ 2048

typedef __attribute__((ext_vector_type(2))) float v2f;
typedef __attribute__((ext_vector_type(8))) float v8f;

// ---------------------------------------------------------------------------
// tanh: gfx1250 has V_TANH_F32 (trans op). Use the builtin if exposed,
// otherwise an exp-based fallback (v_exp_f32 + v_rcp_f32).
// ---------------------------------------------------------------------------
__device__ __forceinline__ float fast_tanh(float x) {
#if __has_builtin(__builtin_amdgcn_tanhf)
    return __builtin_amdgcn_tanhf(x);
#else
    float e = __expf(2.0f * x);
    return 1.0f - 2.0f * __builtin_amdgcn_rcpf(e + 1.0f);
#endif
}

// ---------------------------------------------------------------------------
// Kernel 1: xbar[t][e] = (1/B) * sum_b embed[byte[b,t]][e]
// grid: S/8 blocks of 256 (8 t's per block, 32 e-lanes each)
// ---------------------------------------------------------------------------
__global__ void k_xbar(const int* __restrict__ bs, const float* __restrict__ embed,
                       float* __restrict__ xbar) {
    int e = threadIdx.x & 31;
    int t = blockIdx.x * 8 + (threadIdx.x >> 5);
    float acc = 0.0f;
    for (int b = 0; b < B_; ++b) {
        int byte = bs[(size_t)b * S_ + t];           // broadcast across the whole wave
        acc += embed[byte * E_ + e];
    }
    xbar[t * E_ + e] = acc * (1.0f / (float)B_);
}

// ---------------------------------------------------------------------------
// Kernel 2: sum_x[b][e] = sum_t embed[byte[b,t]][e]   (for msum)
// grid: B blocks of 256 (8 waves split the t-range, LDS combine)
// ---------------------------------------------------------------------------
__global__ void k_sumx(const int* __restrict__ bs, const float* __restrict__ embed,
                       float* __restrict__ sum_x) {
    __shared__ float part[8][E_];
    int wave = threadIdx.x >> 5;
    int e = threadIdx.x & 31;
    int b = blockIdx.x;
    float acc = 0.0f;
    const int t0 = wave * (S_ / 8), t1 = t0 + (S_ / 8);
    for (int t = t0; t < t1; ++t) {
        int byte = bs[(size_t)b * S_ + t];
        acc += embed[byte * E_ + e];
    }
    part[wave][e] = acc;
    __syncthreads();
    if (threadIdx.x < E_) {
        float s = 0.0f;
        for (int w = 0; w < 8; ++w) s += part[w][threadIdx.x];   // fixed order
        sum_x[b * E_ + threadIdx.x] = s;
    }
}

// ---------------------------------------------------------------------------
// Kernel 3: the sequential st/lt recurrence. One wave, lane == e.
// Emits v_t = 0.3*st_t + 0.1*lt_t (carry-in values, as in the reference)
// and vsum = sum_t v_t.
// ---------------------------------------------------------------------------
__global__ void k_scan(const float* __restrict__ xbar, float* __restrict__ vmat,
                       float* __restrict__ vsum) {
    int e = threadIdx.x;          // 0..31
    float st = 0.0f, lt = 0.0f, vs = 0.0f;
    for (int t = 0; t < S_; ++t) {
        float xb = xbar[t * E_ + e];
        float v = 0.3f * st + 0.1f * lt;      // uses pre-update carry
        vmat[t * E_ + e] = v;
        vs += v;
        float m = 0.6f * xb + v;              // batch-mean of signal
        st = 0.8f * st + 0.2f * m;            // ST_DECAY = 0.8
        lt = 0.999f * lt + 0.001f * m;        // LT_DECAY = 0.999
    }
    vsum[e] = vs;
}

// ---------------------------------------------------------------------------
// Kernel 4: P = embed @ W  (256x32 @ 32x2048) via V_WMMA_F32_16X16X4_F32.
// One 16x16 output tile per wave; 8 K-steps of 4. 2048 tiles / 8 waves/block.
// fp32 WMMA keeps reference precision.
// A 16x4 layout: lane<16 -> M=lane, VGPR j -> K=j ; lane>=16 -> M=lane-16, K=j+2
// B 4x16 layout: lane<16 -> N=lane, VGPR j -> K=j ; lane>=16 -> N=lane-16, K=j+2
// C/D: VGPR r, lane<16 -> M=r, N=lane ; lane>=16 -> M=r+8, N=lane-16
// ---------------------------------------------------------------------------
__global__ void k_gemm_P(const float* __restrict__ embed, const float* __restrict__ W,
                         float* __restrict__ P) {
    int wave = threadIdx.x >> 5;
    int lane = threadIdx.x & 31;
    int tile = blockIdx.x * 8 + wave;        // 0..2047
    int m0 = (tile >> 7) << 4;               // 16 M-tiles
    int n0 = (tile & 127) << 4;              // 128 N-tiles
    int half = lane >> 4;                    // 0 or 1
    int l = lane & 15;

    v8f c = {};
#pragma unroll
    for (int k0 = 0; k0 < E_; k0 += 4) {
        int ka = k0 + half * 2;
        v2f a, b;
        a.x = embed[(m0 + l) * E_ + ka];
        a.y = embed[(m0 + l) * E_ + ka + 1];
        b.x = W[(size_t)ka * H_ + n0 + l];
        b.y = W[(size_t)(ka + 1) * H_ + n0 + l];
        c = __builtin_amdgcn_wmma_f32_16x16x4_f32(false, a, false, b, (short)0, c,
                                                  false, false);
    }
#pragma unroll
    for (int r = 0; r < 8; ++r) {
        int M = r + half * 8;
        P[(size_t)(m0 + M) * H_ + n0 + l] = c[r];
    }
}

// ---------------------------------------------------------------------------
// Kernel 5 (heavy): one block per time step t.
//   c_t[h] = sum_e v_t[e] * W[e][h]          (2048-wide, into LDS)
//   act[b][h] = tanh(0.6 * P[byte[b,t]][h] + c_t[h])
//   accumulate sum / sumsq  -> std_arr[t] = unbiased std over B*H values
//   t == S-1 additionally writes last_act into d_out.
// ---------------------------------------------------------------------------
__global__ void __launch_bounds__(256) k_step(const int* __restrict__ bs,
                                              const float* __restrict__ P,
                                              const float* __restrict__ vmat,
                                              const float* __restrict__ W,
                                              float* __restrict__ std_arr,
                                              float* __restrict__ last_act) {
    __shared__ float c_s[H_];
    __shared__ float vt_s[E_];
    __shared__ int idx_s[B_];
    __shared__ float ws_s[8], wq_s[8];

    const int tid = threadIdx.x;
    const int t = blockIdx.x;

    if (tid < E_) vt_s[tid] = vmat[t * E_ + tid];
    if (tid < B_) idx_s[tid] = bs[(size_t)tid * S_ + t];
    __syncthreads();

    // c_t = v_t @ W   (32 FMAs per h; W is L2-resident, coalesced reads)
    for (int h = tid; h < H_; h += 256) {
        float acc = 0.0f;
#pragma unroll
        for (int e = 0; e < E_; ++e) acc += vt_s[e] * W[(size_t)e * H_ + h];
        c_s[h] = acc;
    }
    __syncthreads();

    const bool is_last = (t == S_ - 1);
    float s = 0.0f, q = 0.0f;
    for (int b = 0; b < B_; ++b) {
        const float* __restrict__ Prow = P + (size_t)idx_s[b] * H_;
        if (b + 1 < B_) {
            // hint the next gathered row toward the caches (global_prefetch_b8)
            __builtin_prefetch(P + (size_t)idx_s[b + 1] * H_ + tid * 8, 0, 1);
        }
#pragma unroll 4
        for (int h = tid; h < H_; h += 256) {
            float v = fast_tanh(0.6f * Prow[h] + c_s[h]);
            s += v;
            q += v * v;
            if (is_last) last_act[(size_t)b * H_ + h] = v;
        }
    }

    // wave32 reduce (fixed order), then 8 wave-partials combined in double
#pragma unroll
    for (int off = 16; off > 0; off >>= 1) {
        s += __shfl_down(s, off);
        q += __shfl_down(q, off);
    }
    if ((tid & 31) == 0) { ws_s[tid >> 5] = s; wq_s[tid >> 5] = q; }
    __syncthreads();
    if (tid == 0) {
        double S = 0.0, Q = 0.0;
        for (int w = 0; w < 8; ++w) { S += ws_s[w]; Q += wq_s[w]; }
        const double N = (double)B_ * (double)H_;
        double var = (Q - S * S / N) / (N - 1.0);   // ddof=1 (torch.std unbiased)
        std_arr[t] = (float)sqrt(var > 0.0 ? var : 0.0);
    }
}

// ---------------------------------------------------------------------------
// Kernel 6: finalize. esum/S (fixed-order tree reduce over std_arr) and
// msum/S = (0.6*sum_x[b] + vsum) / S.
// ---------------------------------------------------------------------------
__global__ void k_final(const float* __restrict__ std_arr, const float* __restrict__ sum_x,
                        const float* __restrict__ vsum, float* __restrict__ out) {
    __shared__ float red[256];
    const int tid = threadIdx.x;
    float s = 0.0f;
    for (int i = tid; i < S_; i += 256) s += std_arr[i];
    red[tid] = s;
    __syncthreads();
#pragma unroll
    for (int off = 128; off > 0; off >>= 1) {
        if (tid < off) red[tid] += red[tid + off];
        __syncthreads();
    }
    if (tid == 0) out[(size_t)B_ * H_] = red[0] * (1.0f / (float)S_);
    for (int i = tid; i < B_ * E_; i += 256) {
        int e = i & (E_ - 1);
        out[(size_t)B_ * H_ + 1 + i] = (0.6f * sum_x[i] + vsum[e]) * (1.0f / (float)S_);
    }
}

// ---------------------------------------------------------------------------
// Launch. d_out = [ last_act (B*H) | esum/S (1) | msum/S (B*E) ] fp32.
// Scratch layout (~4.1 MB): xbar | vmat | P | sum_x | vsum | std_arr
// ---------------------------------------------------------------------------
extern "C" void kernel_launch(void* const* d_in, const int* in_sizes, int n_in,
                              void* d_out, int out_size, void* d_ws, size_t ws_size,
                              hipStream_t stream) {
    (void)in_sizes; (void)n_in; (void)out_size; (void)ws_size;
    const int*   bs    = (const int*)d_in[0];     // [B, S] int32
    const float* embed = (const float*)d_in[1];   // [256, E]
    const float* W     = (const float*)d_in[2];   // [E, H]
    float* out = (float*)d_out;

    char* ws = (char*)d_ws;
    float* xbar    = (float*)(ws);                                            // S*E
    float* vmat    = (float*)(ws + (size_t)S_ * E_ * 4);                      // S*E
    float* P       = (float*)(ws + (size_t)2 * S_ * E_ * 4);                  // 256*H
    float* sum_x   = (float*)(ws + (size_t)2 * S_ * E_ * 4 + (size_t)256 * H_ * 4);
    float* vsum    = sum_x + B_ * E_;
    float* std_arr = vsum + E_;

    k_xbar  <<<S_ / 8, 256, 0, stream>>>(bs, embed, xbar);
    k_sumx  <<<B_,     256, 0, stream>>>(bs, embed, sum_x);
    k_scan  <<<1,      32,  0, stream>>>(xbar, vmat, vsum);
    k_gemm_P<<<256,    256, 0, stream>>>(embed, W, P);
    k_step  <<<S_,     256, 0, stream>>>(bs, P, vmat, W, std_arr, out);
    k_final <<<1,      256, 0, stream>>>(std_arr, sum_x, vsum, out);
}